// Focal_Loss_29540785062106
// MI455X (gfx1250) — compile-verified
//
#include <hip/hip_runtime.h>
#include <hip/hip_bf16.h>

// Problem constants (from reference setup_inputs)
#define B_IMG 8
#define P_PRI 67995
#define C_CLS 80
#define M_BOX 64
#define QV    (C_CLS / 4)          // 20 float4 per prior row of pred_cls
#define ALPHA_F 0.25f
#define EPS_F   1e-4f
#define ACC_STRIDE 32              // floats per image accumulator row (128 B)
#define CLS_UNROLL 4               // float4 tiles per thread in focal kernel

typedef __attribute__((ext_vector_type(2))) float v2f;
typedef __attribute__((ext_vector_type(8))) float v8f;

// ---------------------------------------------------------------------------
// Wave32 full reduction via V_WMMA_F32_16X16X4_F32 with a ones B-matrix.
// A (16x4 f32): lane L<16 holds A[L][K0],A[L][K1]; lane L>=16 holds A[L-16][K2],A[L-16][K3].
// Partial in component 0, zero in component 1 => row m sum = v(lane m) + v(lane m+16).
// D[m][n] = rowsum(m) for every n; lane 0 holds D[0..7][0], lane 16 holds D[8..15][0].
// Total = (lane0 Σd) + (lane16 Σd). EXEC must be all ones: uniform control flow only.
// ---------------------------------------------------------------------------
__device__ __forceinline__ float wave_sum_wmma(float v) {
  v2f a; a[0] = v;    a[1] = 0.0f;
  v2f b; b[0] = 1.0f; b[1] = 1.0f;
  v8f c = {};
  v8f d = __builtin_amdgcn_wmma_f32_16x16x4_f32(
      /*neg_a=*/false, a, /*neg_b=*/false, b,
      /*c_mod=*/(short)0, c, /*reuse_a=*/false, /*reuse_b=*/false);
  float s = d[0] + d[1] + d[2] + d[3] + d[4] + d[5] + d[6] + d[7];
  return __shfl(s, 0, 32) + __shfl(s, 16, 32);
}

__device__ __forceinline__ float block_sum(float v, float* red /* >= nwaves floats */) {
  float w = wave_sum_wmma(v);
  __syncthreads();
  if ((threadIdx.x & 31) == 0) red[threadIdx.x >> 5] = w;
  __syncthreads();
  float t = 0.0f;
  const int nw = (int)(blockDim.x >> 5);
  for (int i = 0; i < nw; ++i) t += red[i];
  __syncthreads();
  return t;
}

__device__ __forceinline__ float smooth_l1(float d) {
  float ad = fabsf(d);
  return (ad < 1.0f) ? 0.5f * d * d : ad - 0.5f;
}

// ---------------------------------------------------------------------------
// Kernel 0: zero the padded accumulator block (B_IMG rows x 128 B).
// ---------------------------------------------------------------------------
__global__ void focal_zero_kernel(float* __restrict__ acc) {
  const int i = blockIdx.x * blockDim.x + threadIdx.x;
  if (i < B_IMG * ACC_STRIDE) acc[i] = 0.0f;
}

// ---------------------------------------------------------------------------
// Kernel 1: per-prior IoU match + smooth-L1 (positives only) + npos.
// grid = (ceil(P/256), B). Boxes/labels/areas staged in LDS.
// ---------------------------------------------------------------------------
__global__ void focal_match_kernel(const float* __restrict__ pred_loc,
                                   const float* __restrict__ b_boxes,
                                   const int*   __restrict__ b_labels,
                                   const float* __restrict__ priors,
                                   int*   __restrict__ match_info,
                                   float* __restrict__ acc) {
  __shared__ float4 s_box[M_BOX];
  __shared__ float  s_area[M_BOX];
  __shared__ int    s_lab[M_BOX];
  __shared__ float  s_red[8];

  const int b = blockIdx.y;
  const int p = blockIdx.x * blockDim.x + threadIdx.x;

  if (threadIdx.x < M_BOX) {
    const int m = threadIdx.x;
    float4 bx = ((const float4*)b_boxes)[b * M_BOX + m];
    s_box[m]  = bx;
    s_area[m] = (bx.z - bx.x) * (bx.w - bx.y);
    s_lab[m]  = b_labels[b * M_BOX + m];
  }
  __syncthreads();

  const bool valid = (p < P_PRI);
  float sl1 = 0.0f, pos_f = 0.0f;

  if (valid) {
    float4 pr = ((const float4*)priors)[p];          // cx, cy, w, h
    const float px0 = pr.x - pr.z * 0.5f;
    const float py0 = pr.y - pr.w * 0.5f;
    const float px1 = pr.x + pr.z * 0.5f;
    const float py1 = pr.y + pr.w * 0.5f;
    const float areaP = (px1 - px0) * (py1 - py0);

    float best = -1.0f;
    int   bidx = 0;
    #pragma unroll 8
    for (int m = 0; m < M_BOX; ++m) {
      float4 bx = s_box[m];
      float lx = fmaxf(px0, bx.x), ly = fmaxf(py0, bx.y);
      float rx = fminf(px1, bx.z), ry = fminf(py1, bx.w);
      float w  = fmaxf(rx - lx, 0.0f);
      float h  = fmaxf(ry - ly, 0.0f);
      float inter = w * h;
      float uni   = fmaxf(areaP + s_area[m] - inter, 1e-10f);
      float iou   = inter / uni;
      if (iou > best) { best = iou; bidx = m; }      // strict > keeps first max
    }

    const bool positive = (best >= 0.5f);
    const bool negative = (best < 0.4f);
    const int info = (s_lab[bidx] & 0xFF) | (positive ? 0x100 : 0) | (negative ? 0x200 : 0);
    match_info[b * P_PRI + p] = info;

    if (positive) {
      pos_f = 1.0f;
      float4 bx = s_box[bidx];
      const float gcx = (bx.x + bx.z) * 0.5f;
      const float gcy = (bx.y + bx.w) * 0.5f;
      const float gw  = bx.z - bx.x;
      const float gh  = bx.w - bx.y;
      // encode with variances (0.1 center, 0.2 size)
      const float tx = (gcx - pr.x) / (pr.z * 0.1f);
      const float ty = (gcy - pr.y) / (pr.w * 0.1f);
      const float tw = __logf(fmaxf(gw, 1e-10f) / pr.z) / 0.2f;
      const float th = __logf(fmaxf(gh, 1e-10f) / pr.w) / 0.2f;
      float4 lp = ((const float4*)pred_loc)[b * P_PRI + p];
      sl1 = smooth_l1(lp.x - tx) + smooth_l1(lp.y - ty) +
            smooth_l1(lp.z - tw) + smooth_l1(lp.w - th);
    }
  }

  const float tot_sl1 = block_sum(sl1, s_red);
  const float tot_pos = block_sum(pos_f, s_red);
  if (threadIdx.x == 0) {
    atomicAdd(&acc[b * ACC_STRIDE + 1], tot_sl1);
    atomicAdd(&acc[b * ACC_STRIDE + 2], tot_pos);
  }
}

// ---------------------------------------------------------------------------
// Kernel 2: focal classification loss. Each thread streams CLS_UNROLL coalesced
// float4 tiles of pred_cls (the bandwidth-dominant 174 MB), prefetching the
// next tile (global_prefetch_b8). 4x coarsening cuts per-image atomics to
// ~1.3K, and accumulators are padded to independent 128 B cachelines.
// grid = (ceil(P*QV/(256*CLS_UNROLL)), B).
// ---------------------------------------------------------------------------
__global__ void focal_cls_kernel(const float* __restrict__ pred_cls,
                                 const int*   __restrict__ match_info,
                                 float* __restrict__ acc) {
  __shared__ float s_red[8];
  const int b  = blockIdx.y;
  const int NV = P_PRI * QV;
  const float4* __restrict__ cls4 = (const float4*)pred_cls + (size_t)b * NV;
  const int base = blockIdx.x * (blockDim.x * CLS_UNROLL) + threadIdx.x;

  float loss = 0.0f;
  #pragma unroll
  for (int u = 0; u < CLS_UNROLL; ++u) {
    const int idx = base + u * blockDim.x;
    if (idx < NV) {
      // Hint-prefetch the next unroll step's cacheline (speculative, RT).
      if (u + 1 < CLS_UNROLL) {
        __builtin_prefetch((const void*)(cls4 + (idx + blockDim.x)), 0, 0);
      }
      const int p = idx / QV;
      const int q = idx - p * QV;
      float4 cp = cls4[idx];
      const int info   = match_info[b * P_PRI + p];
      const int label  = info & 0xFF;
      const bool pos   = (info & 0x100) != 0;
      const bool count = (info & 0x300) != 0;   // positive or negative (not ignore)
      if (count) {
        float vals[4] = {cp.x, cp.y, cp.z, cp.w};
        #pragma unroll
        for (int j = 0; j < 4; ++j) {
          const int  c      = q * 4 + j;
          const bool is_one = pos && (c == label);
          float pv  = fminf(fmaxf(vals[j], EPS_F), 1.0f - EPS_F);
          float a_t = is_one ? ALPHA_F : (1.0f - ALPHA_F);
          float p_t = is_one ? pv : (1.0f - pv);
          float om  = 1.0f - p_t;
          loss += a_t * om * om * (-__logf(p_t));
        }
      }
    }
  }

  const float tot = block_sum(loss, s_red);
  if (threadIdx.x == 0) atomicAdd(&acc[b * ACC_STRIDE + 0], tot);
}

// ---------------------------------------------------------------------------
// Kernel 3: finalize -> out = [total, loc, conf] (reference flat return order)
// ---------------------------------------------------------------------------
__global__ void focal_finalize_kernel(const float* __restrict__ acc,
                                      float* __restrict__ out) {
  if (threadIdx.x == 0 && blockIdx.x == 0) {
    float conf = 0.0f, sl1 = 0.0f, np = 0.0f;
    for (int b = 0; b < B_IMG; ++b) {
      const float cls_b  = acc[b * ACC_STRIDE + 0];
      const float sl1_b  = acc[b * ACC_STRIDE + 1];
      const float npos_b = acc[b * ACC_STRIDE + 2];
      conf += cls_b / fmaxf(npos_b, 1.0f);
      sl1  += sl1_b;
      np   += npos_b;
    }
    conf *= (1.0f / (float)B_IMG);
    const float loc = sl1 / fmaxf(np, 1.0f);
    out[0] = conf + loc;   // total_loss
    out[1] = loc;          // loc_loss
    out[2] = conf;         // conf_loss
  }
}

// ---------------------------------------------------------------------------
extern "C" void kernel_launch(void* const* d_in, const int* in_sizes, int n_in,
                              void* d_out, int out_size, void* d_ws, size_t ws_size,
                              hipStream_t stream) {
  const float* pred_loc = (const float*)d_in[0];   // (B,P,4)
  const float* pred_cls = (const float*)d_in[1];   // (B,P,C)
  const float* b_boxes  = (const float*)d_in[2];   // (B,M,4)
  const int*   b_labels = (const int*)  d_in[3];   // (B,M)
  const float* priors   = (const float*)d_in[4];   // (P,4)
  float* out = (float*)d_out;

  // Workspace: [acc: B_IMG x 32 floats (128 B/image: cls,sl1,npos,...) | match int[B*P]]
  float* acc   = (float*)d_ws;
  int*   match = (int*)((char*)d_ws + B_IMG * ACC_STRIDE * sizeof(float));

  focal_zero_kernel<<<1, B_IMG * ACC_STRIDE, 0, stream>>>(acc);

  {
    dim3 grid((P_PRI + 255) / 256, B_IMG);
    focal_match_kernel<<<grid, 256, 0, stream>>>(pred_loc, b_boxes, b_labels,
                                                 priors, match, acc);
  }
  {
    const int NV = P_PRI * QV;
    const int tile = 256 * CLS_UNROLL;
    dim3 grid((NV + tile - 1) / tile, B_IMG);
    focal_cls_kernel<<<grid, 256, 0, stream>>>(pred_cls, match, acc);
  }
  focal_finalize_kernel<<<1, 32, 0, stream>>>(acc, out);
}